// Feature_69045894250876
// MI455X (gfx1250) — compile-verified
//
#include <hip/hip_runtime.h>
#include <math.h>

typedef __attribute__((ext_vector_type(2))) float v2f;
typedef __attribute__((ext_vector_type(8))) float v8f;

#define NTYPES 6
#define NN     96
#define NPAIR  95
#define L0     25
#define L1     50
#define L2     100
#define AX     4

// ---- LDS layout (float offsets inside s_big) ----
// phase 1: h0 [96][25] @0
//          Wcat1 pair-interleaved [6][14 kp][50 o][2] @2400 (k rows 25..27/t zero)
//          h1 [96][51] @10800
// phase 2: W2-stage pair-interleaved [26 kp][100 o][2] @0 (k rows 50,51 zero)
// phase 3: h2 pair-interleaved [48 mp][101... actually [mp][o][2] stride 202] @0,
//          t1 [12] @9696, t2 pair-interleaved [48 mp][4 d][2] @9712
#define OFF_H0 0
#define OFF_W1 2400
#define OFF_H1 10800
#define H1S    51
#define OFF_T1 9696
#define OFF_T2 9712
#define BIG_FLOATS 15696   // 62784 bytes (+2304 B geometry = 65088 <= 64 KiB)

__device__ __forceinline__ float fast_tanh(float x) {
    // robust: x->+inf => 1, x->-inf => -1, exact 0 at 0, NaN propagates
    const float e = __expf(2.0f * x);
    return 1.0f - 2.0f / (e + 1.0f);
}

__global__ __launch_bounds__(192)
void pairmlp_fused(const float* __restrict__ coords,
                   const int*   __restrict__ atom_types,
                   const unsigned char* __restrict__ qm_mask,   // jax bool -> 1 byte
                   const float* __restrict__ W0, const float* __restrict__ b0,
                   const float* __restrict__ W1, const float* __restrict__ b1,
                   const float* __restrict__ W2, const float* __restrict__ b2,
                   float* __restrict__ out)
{
    __shared__ float s_a[NN * 3];
    __shared__ float s_pm[NN];
    __shared__ int   s_tj[NN];
    __shared__ float s_dinv[NN];
    __shared__ __align__(16) float s_big[BIG_FLOATS];

    const int tid  = threadIdx.x;
    const int lane = tid & 31;
    const int wave = tid >> 5;          // 0..5, M-tile owner
    const int row  = blockIdx.x;        // b*96 + n
    const int bb   = row / NN;
    const int n    = row % NN;

    const int   ti = atom_types[bb * NN + n];
    const float qi = qm_mask[bb * NN + n] ? 1.0f : 0.0f;

    // ---------------- phase 0: pair geometry (pairs padded 95 -> 96) ----------
    if (tid < NN) {
        const int p = tid;
        if (p < NPAIR) {
            const int j  = p + (p >= n ? 1 : 0);
            const float dx = coords[(bb*NN + n)*3 + 0] - coords[(bb*NN + j)*3 + 0];
            const float dy = coords[(bb*NN + n)*3 + 1] - coords[(bb*NN + j)*3 + 1];
            const float dz = coords[(bb*NN + n)*3 + 2] - coords[(bb*NN + j)*3 + 2];
            const float d    = sqrtf(dx*dx + dy*dy + dz*dz);
            const float dinv = 1.0f / fmaxf(d, 1e-12f);
            const float w    = dinv * dinv;
            s_a[p*3+0] = dx * w;  s_a[p*3+1] = dy * w;  s_a[p*3+2] = dz * w;
            s_dinv[p]  = dinv;
            s_tj[p]    = atom_types[bb*NN + j];
            s_pm[p]    = (qm_mask[bb*NN + n] && qm_mask[bb*NN + j]) ? 1.0f : 0.0f;
        } else {
            s_a[p*3+0] = 0.f; s_a[p*3+1] = 0.f; s_a[p*3+2] = 0.f;
            s_dinv[p]  = 0.f; s_tj[p] = 0; s_pm[p] = 0.f;
        }
    }
    __syncthreads();

    // ---------------- layer 0 (elementwise): h0 = tanh(W0*dinv + b0) ----------
    for (int idx = tid; idx < NN * L0; idx += 192) {
        const int p = idx / L0, o = idx % L0;
        const int ch = ti * NTYPES + s_tj[p];
        s_big[OFF_H0 + p*L0 + o] = fast_tanh(W0[ch*L0 + o] * s_dinv[p] + b0[ch*L0 + o]);
    }
    // stage Wcat1 pair-interleaved: (t,k,o) -> ((t*14 + k/2)*50 + o)*2 + (k&1)
    for (int idx = tid; idx < NTYPES * 14 * L1 * 2; idx += 192) {
        const int t   = idx / (14 * L1 * 2);
        const int rem = idx % (14 * L1 * 2);
        const int kp  = rem / (L1 * 2);
        const int o   = (rem >> 1) % L1;
        const int k   = kp*2 + (rem & 1);
        s_big[OFF_W1 + idx] =
            (k < L0) ? W1[((ti*NTYPES + t)*L1 + o)*L0 + k] : 0.0f;
    }
    __syncthreads();

    const v8f vzero = {0.f,0.f,0.f,0.f,0.f,0.f,0.f,0.f};
    const int am    = wave*16 + (lane & 15);    // A-fragment pair row
    const int cc    = lane & 15;                // fragment column
    const int koff  = (lane >> 4) << 1;         // K sub-offset 0 or 2
    const int kphi  = koff >> 1;                // K pair sub-offset 0 or 1
    const int tjm   = s_tj[am];

    // ---------------- layer 1: type-expanded GEMM, K=6*28, N=50 ----------------
    float a1x[7], a1y[7];
    {
        const float* h0r = &s_big[OFF_H0 + am * L0];
#pragma unroll
        for (int ks = 0; ks < 6; ++ks) {
            a1x[ks] = h0r[ks*4 + koff];
            a1y[ks] = h0r[ks*4 + koff + 1];
        }
        a1x[6] = h0r[24];   // only k=24 feeds a nonzero B row in the tail step
        a1y[6] = a1x[6];
    }

    v8f acc1[4];
#pragma unroll
    for (int nt = 0; nt < 4; ++nt) acc1[nt] = vzero;

    const v2f* w1p = reinterpret_cast<const v2f*>(&s_big[OFF_W1]);
    for (int t = 0; t < NTYPES; ++t) {
        const bool sel = (tjm == t);
#pragma unroll
        for (int ks = 0; ks < 7; ++ks) {
            v2f a;
            a.x = sel ? a1x[ks] : 0.0f;
            a.y = sel ? a1y[ks] : 0.0f;
            const v2f* wb = w1p + (t*14 + ks*2 + kphi)*L1 + cc;
#pragma unroll
            for (int nt = 0; nt < 4; ++nt) {
                const v2f b = wb[nt*16];     // single ds_load_b64, cols>=50 discarded
                acc1[nt] = __builtin_amdgcn_wmma_f32_16x16x4_f32(
                    false, a, false, b, (short)0, acc1[nt], false, false);
            }
        }
    }
    // epilogue: h1 = tanh(y + b1) + [h0,h0]
#pragma unroll
    for (int nt = 0; nt < 4; ++nt) {
        const int o = nt*16 + cc;
        if (o < L1) {
#pragma unroll
            for (int r = 0; r < 8; ++r) {
                const int m  = wave*16 + r + ((lane >> 4) << 3);
                const int ch = ti*NTYPES + s_tj[m];
                const float y = acc1[nt][r] + b1[ch*L1 + o];
                s_big[OFF_H1 + m*H1S + o] = fast_tanh(y) + s_big[OFF_H0 + m*L0 + (o % L0)];
            }
        }
    }

    // ---------------- layer 2: stream W2 per neighbor-type, K=52, N=100 --------
    float a2x[13], a2y[13];
    {
        const float* h1r = &s_big[OFF_H1 + am * H1S];
#pragma unroll
        for (int ks = 0; ks < 12; ++ks) {
            a2x[ks] = h1r[ks*4 + koff];
            a2y[ks] = h1r[ks*4 + koff + 1];
        }
        a2x[12] = h1r[48];   // B rows 50,51 are zero in the tail step
        a2y[12] = h1r[49];
    }

    v8f acc2[7];
#pragma unroll
    for (int nt = 0; nt < 7; ++nt) acc2[nt] = vzero;

    const v2f* w2p = reinterpret_cast<const v2f*>(&s_big[0]);
    for (int t = 0; t < NTYPES; ++t) {
        __syncthreads();   // protect staging buffer (and h0/Wcat1 region on t==0)
        // stage W2 pair-interleaved: (k,o) -> ((k/2)*100 + o)*2 + (k&1)
        for (int idx = tid; idx < 26 * L2 * 2; idx += 192) {
            const int kp = idx / (L2 * 2);
            const int o  = (idx >> 1) % L2;
            const int k  = kp*2 + (idx & 1);
            s_big[idx] = (k < L1) ? W2[((ti*NTYPES + t)*L2 + o)*L1 + k] : 0.0f;
        }
        __syncthreads();
        const bool sel = (tjm == t);
#pragma unroll
        for (int ks = 0; ks < 13; ++ks) {
            v2f a;
            a.x = sel ? a2x[ks] : 0.0f;
            a.y = sel ? a2y[ks] : 0.0f;
            const v2f* wb = w2p + (ks*2 + kphi)*L2 + cc;
#pragma unroll
            for (int nt = 0; nt < 7; ++nt) {
                const v2f b = wb[nt*16];     // single ds_load_b64, cols>=100 discarded
                acc2[nt] = __builtin_amdgcn_wmma_f32_16x16x4_f32(
                    false, a, false, b, (short)0, acc2[nt], false, false);
            }
        }
    }
    // epilogue in registers: h2 = (tanh(y+b2) + [h1,h1]) * pair_mask
#pragma unroll
    for (int nt = 0; nt < 7; ++nt) {
        const int o = nt*16 + cc;
#pragma unroll
        for (int r = 0; r < 8; ++r) {
            const int m = wave*16 + r + ((lane >> 4) << 3);
            float v = 0.0f;
            if (o < L2) {
                const int ch = ti*NTYPES + s_tj[m];
                const float y = acc2[nt][r] + b2[ch*L2 + o];
                v = (fast_tanh(y) + s_big[OFF_H1 + m*H1S + (o % L1)]) * s_pm[m];
            }
            acc2[nt][r] = v;
        }
    }
    __syncthreads();   // all waves done reading W2 stage
    // store h2 pair-interleaved: (m,o) -> (m/2)*202 + o*2 + (m&1)
#pragma unroll
    for (int nt = 0; nt < 7; ++nt) {
        const int o = nt*16 + cc;
        if (o < L2) {
#pragma unroll
            for (int r = 0; r < 8; ++r) {
                const int m = wave*16 + r + ((lane >> 4) << 3);
                s_big[(m >> 1)*202 + o*2 + (m & 1)] = acc2[nt][r];
            }
        }
    }
    __syncthreads();

    // ---------------- t1 (3x4) and t2 (96x4), small VALU reductions ------------
    if (tid < 12) {
        const int c = tid >> 2, d = tid & 3;
        float s = 0.0f;
        for (int m = 0; m < NN; ++m)
            s += s_a[m*3 + c] * s_pm[m] * s_big[(m >> 1)*202 + d*2 + (m & 1)];
        s_big[OFF_T1 + tid] = s;
    }
    __syncthreads();
    // t2 pair-interleaved: (m,d) -> OFF_T2 + (m/2)*8 + d*2 + (m&1)
    for (int idx = tid; idx < NN * AX; idx += 192) {
        const int m = idx >> 2, d = idx & 3;
        float s = 0.0f;
        for (int c = 0; c < 3; ++c)
            s += s_a[m*3 + c] * s_pm[m] * s_big[OFF_T1 + c*AX + d];
        s_big[OFF_T2 + (m >> 1)*8 + d*2 + (m & 1)] = s;
    }
    __syncthreads();

    // ---------------- res: resT(4x100) = t2^T(4x96) x h2(96x100) via WMMA ------
    // A rows d>=4 and B cols o>=100 read harmless garbage (discarded).
    {
        v8f accA = vzero, accB = vzero;
        const int n0a = wave * 16;           // waves 0..5 -> n-tiles 0..5
        const v2f* hp = reinterpret_cast<const v2f*>(&s_big[0]);       // [mp][o]
        const v2f* tp = reinterpret_cast<const v2f*>(&s_big[OFF_T2]);  // [mp][d]
#pragma unroll
        for (int ks = 0; ks < 24; ++ks) {
            const int mp = ks*2 + kphi;      // m0 = mp*2
            const v2f a = tp[mp*4 + cc];
            const v2f b = hp[mp*101 + n0a + cc];
            accA = __builtin_amdgcn_wmma_f32_16x16x4_f32(
                false, a, false, b, (short)0, accA, false, false);
            if (wave == 0) {                 // wave-uniform: EXEC stays full
                const v2f b2f = hp[mp*101 + 96 + cc];
                accB = __builtin_amdgcn_wmma_f32_16x16x4_f32(
                    false, a, false, b2f, (short)0, accB, false, false);
            }
        }
        const long obase = (long)(bb*NN + n) * L2 * AX;
        if (lane < 16) {                     // D rows 0..3 live in r=0..3, lanes 0..15
            const int oA = n0a + cc;         // < 96 always
#pragma unroll
            for (int r = 0; r < 4; ++r)
                out[obase + oA*AX + r] = accA[r] * qi;
            if (wave == 0 && cc < 4) {       // tile 6: o = 96..99
                const int oB = 96 + cc;
#pragma unroll
                for (int r = 0; r < 4; ++r)
                    out[obase + oB*AX + r] = accB[r] * qi;
            }
        }
    }
}

extern "C" void kernel_launch(void* const* d_in, const int* in_sizes, int n_in,
                              void* d_out, int out_size, void* d_ws, size_t ws_size,
                              hipStream_t stream) {
    const float* coords = (const float*)d_in[0];
    const int*   atypes = (const int*)d_in[1];
    const unsigned char* qm = (const unsigned char*)d_in[2];
    const float* W0 = (const float*)d_in[3];
    const float* b0 = (const float*)d_in[4];
    const float* W1 = (const float*)d_in[5];
    const float* b1 = (const float*)d_in[6];
    const float* W2 = (const float*)d_in[7];
    const float* b2 = (const float*)d_in[8];
    const int rows = in_sizes[1];   // B*N = 192
    pairmlp_fused<<<rows, 192, 0, stream>>>(coords, atypes, qm,
                                            W0, b0, W1, b1, W2, b2,
                                            (float*)d_out);
}